// PillarMaxPooling_8942121910497
// MI455X (gfx1250) — compile-verified
//
#include <hip/hip_runtime.h>
#include <hip/hip_bf16.h>

// ---------------------------------------------------------------------------
// PillarMaxPooling for MI455X (gfx1250, wave32).
// GEMM (Nx11 @ 11x32) done with V_WMMA_F32_16X16X4_F32 tiles:
//   per wave: 16 points x 32 channels, K=11 padded to 12 -> 3 K-steps x 2 halves.
// Pass 1: WMMA -> per-channel sum / sumsq (deterministic tree reduction).
// Pass 2: finalize BN scale/bias per channel.
// Pass 3: WMMA -> y = relu(h*scale+bias) -> atomicMax (uint bits, y>=0) scatter.
// ---------------------------------------------------------------------------

typedef __attribute__((ext_vector_type(2))) float v2f;
typedef __attribute__((ext_vector_type(8))) float v8f;

#define C_MLP 32
#define C_GRP 11
#define STATS_BLOCKS 256
#define BLOCK_THREADS 256
#define WAVES_PER_BLOCK (BLOCK_THREADS / 32)

__device__ __forceinline__ float wget(const float* __restrict__ W1, int k, int c) {
  // W1 is (11,32) row-major; pad K to 12 with zeros.
  return (k < C_GRP) ? W1[k * C_MLP + c] : 0.0f;
}

// Load the loop-invariant B fragments for this lane.
// B (4x16) layout assumption mirrors C/D: VGPR r, lanes0-15 -> K=r, lanes16-31 -> K=r+2.
__device__ __forceinline__ void load_b_frags(const float* __restrict__ W1, int lane,
                                             v2f bLo[3], v2f bHi[3]) {
  int ln = lane & 15;
  int hi = lane >> 4;
#pragma unroll
  for (int s = 0; s < 3; ++s) {
    int k0 = 4 * s + 2 * hi;
    bLo[s].x = wget(W1, k0,     ln);
    bLo[s].y = wget(W1, k0 + 1, ln);
    bHi[s].x = wget(W1, k0,     16 + ln);
    bHi[s].y = wget(W1, k0 + 1, 16 + ln);
  }
}

// Compute one 16-point tile: h = G @ W1 via 6x V_WMMA_F32_16X16X4_F32.
// cLo = channels 0..15, cHi = channels 16..31. sid_out = pillar id of point base+(lane&15).
__device__ __forceinline__ void compute_tile(const float* __restrict__ xyz,
                                             const float* __restrict__ ptf,
                                             const int* __restrict__ pidx,
                                             const int* __restrict__ psi,
                                             const v2f bLo[3], const v2f bHi[3],
                                             int base, int N, int lane,
                                             v8f& cLo, v8f& cHi, int& sid_out) {
  int ln = lane & 15;
  int hi = lane >> 4;
  int p  = base + ln;

  float f[12];
#pragma unroll
  for (int i = 0; i < 12; ++i) f[i] = 0.0f;
  int sid = 0;
  if (p < N) {
    sid = psi[p];
    float px = (float)pidx[sid * 3 + 2];
    float py = (float)pidx[sid * 3 + 1];
    float cx = (px + 0.5f) * 0.075f - 54.0f;
    float cy = (py + 0.5f) * 0.075f - 54.0f;
    float x = xyz[p * 3 + 0];
    float y = xyz[p * 3 + 1];
    float z = xyz[p * 3 + 2];
    f[0] = x - cx; f[1] = y - cy; f[2] = z + 1.0f;  // cz = 0.5*(-5+3) = -1
    f[3] = x;      f[4] = y;      f[5] = z;
#pragma unroll
    for (int i = 0; i < 5; ++i) f[6 + i] = ptf[p * 5 + i];
  }

  v8f lo = {};
  v8f hiacc = {};
#pragma unroll
  for (int s = 0; s < 3; ++s) {
    // A-frag (16x4 f32): lanes0-15 hold K=4s,4s+1; lanes16-31 hold K=4s+2,4s+3.
    v2f a;
    a.x = hi ? f[4 * s + 2] : f[4 * s + 0];
    a.y = hi ? f[4 * s + 3] : f[4 * s + 1];
    lo    = __builtin_amdgcn_wmma_f32_16x16x4_f32(false, a, false, bLo[s], (short)0, lo,    false, false);
    hiacc = __builtin_amdgcn_wmma_f32_16x16x4_f32(false, a, false, bHi[s], (short)0, hiacc, false, false);
  }
  cLo = lo;
  cHi = hiacc;
  sid_out = sid;
}

// ---------------- kernel 0: zero the output (scatter-max identity) ----------
__global__ void __launch_bounds__(256) pm_init_out(float* __restrict__ out, int n) {
  int i = blockIdx.x * blockDim.x + threadIdx.x;
  int stride = gridDim.x * blockDim.x;
  for (; i < n; i += stride) out[i] = 0.0f;
}

// ---------------- kernel 1: BN statistics pass (deterministic) --------------
__global__ void __launch_bounds__(BLOCK_THREADS) pm_stats(
    const float* __restrict__ xyz, const float* __restrict__ ptf,
    const float* __restrict__ W1, const int* __restrict__ pidx,
    const int* __restrict__ psi, float* __restrict__ partial, int N) {
  __shared__ float red_s[WAVES_PER_BLOCK][C_MLP];
  __shared__ float red_q[WAVES_PER_BLOCK][C_MLP];

  int lane = threadIdx.x & 31;
  int wave = threadIdx.x >> 5;
  int ln   = lane & 15;

  v2f bLo[3], bHi[3];
  load_b_frags(W1, lane, bLo, bHi);

  int numTiles = (N + 15) >> 4;
  float sLo = 0.0f, qLo = 0.0f, sHi = 0.0f, qHi = 0.0f;

  for (int tile = blockIdx.x * WAVES_PER_BLOCK + wave; tile < numTiles;
       tile += gridDim.x * WAVES_PER_BLOCK) {
    int base = tile << 4;
    v8f cLo, cHi;
    int sid;
    compute_tile(xyz, ptf, pidx, psi, bLo, bHi, base, N, lane, cLo, cHi, sid);
    // Padded (p >= N) rows have zero A-rows -> h == 0 -> contribute nothing.
#pragma unroll
    for (int i = 0; i < 8; ++i) {
      float a = cLo[i];
      float b = cHi[i];
      sLo += a; qLo += a * a;
      sHi += b; qHi += b * b;
    }
  }

  // Combine rows 0-7 (lanes 0-15) with rows 8-15 (lanes 16-31): same channel.
  sLo += __shfl_xor(sLo, 16, 32);
  qLo += __shfl_xor(qLo, 16, 32);
  sHi += __shfl_xor(sHi, 16, 32);
  qHi += __shfl_xor(qHi, 16, 32);

  if (lane < 16) {
    red_s[wave][ln]      = sLo;
    red_q[wave][ln]      = qLo;
    red_s[wave][16 + ln] = sHi;
    red_q[wave][16 + ln] = qHi;
  }
  __syncthreads();

  if (threadIdx.x < C_MLP) {
    int c = threadIdx.x;
    float S = 0.0f, Q = 0.0f;
#pragma unroll
    for (int w = 0; w < WAVES_PER_BLOCK; ++w) {
      S += red_s[w][c];
      Q += red_q[w][c];
    }
    partial[blockIdx.x * 64 + c]      = S;
    partial[blockIdx.x * 64 + 32 + c] = Q;
  }
}

// ---------------- kernel 2: finalize BN scale/bias --------------------------
__global__ void pm_finalize(const float* __restrict__ partial,
                            const float* __restrict__ gamma,
                            const float* __restrict__ beta,
                            float* __restrict__ stats, int N) {
  int c = threadIdx.x;  // 32 threads
  float S = 0.0f, Q = 0.0f;
  for (int b = 0; b < STATS_BLOCKS; ++b) {
    S += partial[b * 64 + c];
    Q += partial[b * 64 + 32 + c];
  }
  float invN  = 1.0f / (float)N;
  float mean  = S * invN;
  float var   = Q * invN - mean * mean;
  float scale = gamma[c] * rsqrtf(var + 1e-3f);
  stats[c]      = scale;
  stats[32 + c] = beta[c] - mean * scale;
}

// ---------------- kernel 3: normalize + ReLU + scatter-max ------------------
__global__ void __launch_bounds__(BLOCK_THREADS) pm_pool(
    const float* __restrict__ xyz, const float* __restrict__ ptf,
    const float* __restrict__ W1, const int* __restrict__ pidx,
    const int* __restrict__ psi, const float* __restrict__ stats,
    float* __restrict__ out, int N) {
  int lane = threadIdx.x & 31;
  int wave = threadIdx.x >> 5;
  int ln   = lane & 15;
  int hi   = lane >> 4;

  v2f bLo[3], bHi[3];
  load_b_frags(W1, lane, bLo, bHi);

  float scLo = stats[ln];
  float biLo = stats[32 + ln];
  float scHi = stats[16 + ln];
  float biHi = stats[48 + ln];

  int numTiles = (N + 15) >> 4;
  for (int tile = blockIdx.x * WAVES_PER_BLOCK + wave; tile < numTiles;
       tile += gridDim.x * WAVES_PER_BLOCK) {
    int base = tile << 4;
    v8f cLo, cHi;
    int sid;
    compute_tile(xyz, ptf, pidx, psi, bLo, bHi, base, N, lane, cLo, cHi, sid);

#pragma unroll
    for (int i = 0; i < 8; ++i) {
      int row = i + 8 * hi;          // matrix row held in this lane's VGPR i
      int p   = base + row;
      int s   = __shfl(sid, row, 32);  // pillar id loaded by lane `row`
      if (p < N) {
        float yLo = fmaxf(cLo[i] * scLo + biLo, 0.0f);
        float yHi = fmaxf(cHi[i] * scHi + biHi, 0.0f);
        // y >= 0 -> float max == unsigned-bit max; exact & order-independent.
        atomicMax((unsigned int*)&out[s * C_MLP + ln],      __float_as_uint(yLo));
        atomicMax((unsigned int*)&out[s * C_MLP + 16 + ln], __float_as_uint(yHi));
      }
    }
  }
}

// ---------------------------------------------------------------------------
extern "C" void kernel_launch(void* const* d_in, const int* in_sizes, int n_in,
                              void* d_out, int out_size, void* d_ws, size_t ws_size,
                              hipStream_t stream) {
  const float* xyz   = (const float*)d_in[0];
  const float* ptf   = (const float*)d_in[1];
  const float* W1    = (const float*)d_in[2];
  const float* gamma = (const float*)d_in[3];
  const float* beta  = (const float*)d_in[4];
  const int*   pidx  = (const int*)d_in[5];
  const int*   psi   = (const int*)d_in[6];
  // d_in[7] = xyz_batch_cnt : unused by the reference math.

  int N = in_sizes[6];        // points
  int M = in_sizes[5] / 3;    // pillars

  float* out     = (float*)d_out;
  float* partial = (float*)d_ws;                  // STATS_BLOCKS * 64 floats
  float* stats   = partial + STATS_BLOCKS * 64;   // 64 floats (scale[32], bias[32])

  int nOut = M * C_MLP;
  pm_init_out<<<(nOut + 255) / 256, 256, 0, stream>>>(out, nOut);
  pm_stats<<<STATS_BLOCKS, BLOCK_THREADS, 0, stream>>>(xyz, ptf, W1, pidx, psi, partial, N);
  pm_finalize<<<1, C_MLP, 0, stream>>>(partial, gamma, beta, stats, N);
  pm_pool<<<1024, BLOCK_THREADS, 0, stream>>>(xyz, ptf, W1, pidx, psi, stats, out, N);
}